// CenterLoss_1271310320184
// MI455X (gfx1250) — compile-verified
//
#include <hip/hip_runtime.h>

#define NUM_CLASSES 100000
#define FEAT_DIM    256
#define BATCH       16384
#define ALPHA_F     0.5f
#define EPS_F       1e-6f

typedef __attribute__((ext_vector_type(2))) float        v2f;
typedef __attribute__((ext_vector_type(8))) float        v8f;
typedef __attribute__((ext_vector_type(4))) unsigned int u32x4;
typedef __attribute__((ext_vector_type(4))) int          i32x4;
typedef __attribute__((ext_vector_type(8))) int          i32x8;

// ---------------------------------------------------------------------------
// Kernel 1: histogram of targets -> float counts in workspace
// ---------------------------------------------------------------------------
__global__ void hist_kernel(const int* __restrict__ target,
                            float* __restrict__ counts) {
    int i = blockIdx.x * blockDim.x + threadIdx.x;
    if (i < BATCH) {
        int t = target[i];
        unsafeAtomicAdd(&counts[t], 1.0f);
    }
}

// ---------------------------------------------------------------------------
// Kernel 2: bulk copy centers -> new_centers via Tensor Data Mover.
// One wave per workgroup; each moves a 64-row x 1KB tile through LDS with
// tensor_load_to_lds / tensor_store_from_lds (TENSORcnt-tracked DMA).
// ---------------------------------------------------------------------------
__global__ void __launch_bounds__(32) copy_centers_tdm(
        const float* __restrict__ src, float* __restrict__ dst, int nrows) {
    __shared__ float lds[64 * FEAT_DIM];          // 64 KB staging tile
    lds[threadIdx.x] = 0.0f;                      // keep LDS allocation live

    int row0 = blockIdx.x * 64;
    int rows = nrows - row0;
    rows = rows > 64 ? 64 : rows;
    if (rows <= 0) return;

    unsigned lds_off = (unsigned)(unsigned long long)&lds[0]; // low 32b = LDS byte offset
    unsigned long long ga_src =
        (unsigned long long)(src + (size_t)row0 * FEAT_DIM);
    unsigned long long ga_dst =
        (unsigned long long)(dst + (size_t)row0 * FEAT_DIM);

    // D# group0: count=1 | lds_addr | global_addr[56:0] | type=2
    u32x4 g0l = {};
    g0l[0] = 1u;
    g0l[1] = lds_off;
    g0l[2] = (unsigned)ga_src;
    g0l[3] = (unsigned)((ga_src >> 32) & 0x01FFFFFFull) | (2u << 30);
    u32x4 g0s = g0l;
    g0s[2] = (unsigned)ga_dst;
    g0s[3] = (unsigned)((ga_dst >> 32) & 0x01FFFFFFull) | (2u << 30);

    // D# group1: data_size=4B; tensor_dim0=256; tensor_dim1=rows;
    //            tile_dim0=256; tile_dim1=rows; tensor_dim0_stride=256
    i32x8 g1 = {};
    g1[0] = (int)(2u << 16);                 // data_size code 2 (=4 bytes)
    g1[1] = (int)((unsigned)FEAT_DIM << 16); // tensor_dim0[15:0] @ bits 63:48
    g1[2] = (int)((unsigned)rows << 16);     // tensor_dim1[15:0] @ bits 95:80
    g1[3] = (int)((unsigned)FEAT_DIM << 16); // tile_dim0 @ bits 127:112
    g1[4] = rows;                            // tile_dim1 (tile_dim2 = 0)
    g1[5] = FEAT_DIM;                        // tensor_dim0_stride low 32
    g1[6] = 0;
    g1[7] = 0;

    i32x4 gz = {};
#if defined(__clang_major__) && (__clang_major__ >= 23)
    i32x8 gz8 = {};
    __builtin_amdgcn_tensor_load_to_lds(g0l, g1, gz, gz, gz8, 0);
    __builtin_amdgcn_s_wait_tensorcnt(0);
    __builtin_amdgcn_tensor_store_from_lds(g0s, g1, gz, gz, gz8, 0);
#else
    __builtin_amdgcn_tensor_load_to_lds(g0l, g1, gz, gz, 0);
    __builtin_amdgcn_s_wait_tensorcnt(0);
    __builtin_amdgcn_tensor_store_from_lds(g0s, g1, gz, gz, 0);
#endif
    __builtin_amdgcn_s_wait_tensorcnt(0);
}

// ---------------------------------------------------------------------------
// Kernel 3: fused gather + MSE loss (via WMMA reduction) + scatter update.
// One wave per sample; lane handles 8 consecutive features.
// Loss: D = A(diff^2) x B(ones) + C with V_WMMA_F32_16X16X4_F32, so every
// column of C holds the row-sums; wave total = col(lane0) + col(lane16).
// ---------------------------------------------------------------------------
__global__ void __launch_bounds__(256) fused_loss_scatter(
        const float* __restrict__ centers,
        const float* __restrict__ features,
        const int*   __restrict__ target,
        const float* __restrict__ counts,
        float* __restrict__ loss_out,
        float* __restrict__ new_centers) {
    int gid    = blockIdx.x * blockDim.x + threadIdx.x;
    int sample = gid >> 5;                 // one wave32 per sample
    int lane   = threadIdx.x & 31;
    if (sample >= BATCH) return;           // uniform per wave -> EXEC all 1s

    int   t   = target[sample];
    float cnt = counts[t];
    float inv = ALPHA_F / (cnt + EPS_F);

    const float4* f4 = (const float4*)(features + (size_t)sample * FEAT_DIM + lane * 8);
    const float4* c4 = (const float4*)(centers  + (size_t)t      * FEAT_DIM + lane * 8);
    float4 f0 = f4[0], f1 = f4[1];
    float4 c0 = c4[0], c1 = c4[1];

    float d[8] = { c0.x - f0.x, c0.y - f0.y, c0.z - f0.z, c0.w - f0.w,
                   c1.x - f1.x, c1.y - f1.y, c1.z - f1.z, c1.w - f1.w };

    // --- WMMA f32 16x16x4 reduction of squared residuals ---
    v8f acc  = {};
    v2f ones = { 1.0f, 1.0f };
#pragma unroll
    for (int p = 0; p < 4; ++p) {
        v2f a;
        a[0] = d[2 * p]     * d[2 * p];
        a[1] = d[2 * p + 1] * d[2 * p + 1];
        acc = __builtin_amdgcn_wmma_f32_16x16x4_f32(
                  false, a, false, ones, (short)0, acc, false, false);
    }

    // --- scatter: new_centers[t] -= alpha * diff / (count + eps) ---
    float* dstrow = new_centers + (size_t)t * FEAT_DIM + lane * 8;
#pragma unroll
    for (int j = 0; j < 8; ++j)
        unsafeAtomicAdd(&dstrow[j], -d[j] * inv);

    // --- finish loss reduction ---
    float s = acc[0] + acc[1] + acc[2] + acc[3] +
              acc[4] + acc[5] + acc[6] + acc[7];
    float wave_sum = __shfl(s, 0, 32) + __shfl(s, 16, 32);
    if (lane == 0)
        unsafeAtomicAdd(loss_out,
                        wave_sum * (1.0f / (float)(BATCH * FEAT_DIM)));
}

// ---------------------------------------------------------------------------
// Launch: memset counts+loss -> histogram -> TDM copy -> fused loss/scatter
// (stream order provides counts->scatter and copy->scatter dependencies)
// ---------------------------------------------------------------------------
extern "C" void kernel_launch(void* const* d_in, const int* in_sizes, int n_in,
                              void* d_out, int out_size, void* d_ws, size_t ws_size,
                              hipStream_t stream) {
    const float* centers  = (const float*)d_in[0];
    const float* features = (const float*)d_in[1];
    const int*   target   = (const int*)d_in[2];
    float* out    = (float*)d_out;     // out[0]=loss, out[1..] = new_centers
    float* counts = (float*)d_ws;      // NUM_CLASSES floats of scratch

    hipMemsetAsync(counts, 0, (size_t)NUM_CLASSES * sizeof(float), stream);
    hipMemsetAsync(out, 0, sizeof(float), stream);

    hist_kernel<<<(BATCH + 255) / 256, 256, 0, stream>>>(target, counts);

    int copy_blocks = (NUM_CLASSES + 63) / 64;
    copy_centers_tdm<<<copy_blocks, 32, 0, stream>>>(centers, out + 1, NUM_CLASSES);

    fused_loss_scatter<<<(BATCH * 32 + 255) / 256, 256, 0, stream>>>(
        centers, features, target, counts, out, out + 1);
}